// GIN_58076547776808
// MI455X (gfx1250) — compile-verified
//
#include <hip/hip_runtime.h>

// GIN forward: 2x (segment-sum aggregate + MLP with fused bias/BN/ReLU)
// N=100000, E=1600000, D_IN=128, D_H=256, D_OUT=64, eps=0 (GIN), BN eps=1e-5.

#define DIN   128
#define DH    256
#define DOUTF 64
#define BN_EPS 1e-5f

typedef float v2f __attribute__((ext_vector_type(2)));
typedef float v8f __attribute__((ext_vector_type(8)));

// ---------------------------------------------------------------------------
// Residual init: acc = feat   (h = (1+eps)*x + agg, eps == 0)
// ---------------------------------------------------------------------------
__global__ __launch_bounds__(256)
void copy_f4(const float4* __restrict__ in, float4* __restrict__ out,
             long long n4) {
  long long i = (long long)blockIdx.x * blockDim.x + threadIdx.x;
  if (i < n4) out[i] = in[i];
}

// ---------------------------------------------------------------------------
// Edge scatter-add: acc[dst[e]][t..t+3] += feat[src[e]][t..t+3]
// One b128 gather + 4 hardware f32 atomics per thread. D is a power of two,
// D/4 >= 32 so the edge index is wave-uniform (broadcast-friendly loads).
// Indices are int64 per the reference (edge_index dtype).
// ---------------------------------------------------------------------------
template<int D>
__global__ __launch_bounds__(256)
void scatter_add4(const float* __restrict__ feat,
                  float* __restrict__ acc,
                  const long long* __restrict__ srcIdx,
                  const long long* __restrict__ dstIdx,
                  long long E) {
  constexpr int Q = D / 4;                       // float4 quads per row
  long long i = (long long)blockIdx.x * blockDim.x + threadIdx.x;
  long long total = E * (long long)Q;
  if (i >= total) return;
  long long e = i / Q;                           // power-of-2 -> shift
  int t = (int)(i - e * (long long)Q) * 4;
  long long s = srcIdx[e];
  long long d = dstIdx[e];
  float4 v = *(const float4*)(feat + s * (long long)D + t);
  float* p = acc + d * (long long)D + t;
  unsafeAtomicAdd(p + 0, v.x);                   // global_atomic_add_f32
  unsafeAtomicAdd(p + 1, v.y);
  unsafeAtomicAdd(p + 2, v.z);
  unsafeAtomicAdd(p + 3, v.w);
}

// ---------------------------------------------------------------------------
// GEMM + bias (+ BN + ReLU) using V_WMMA_F32_16X16X4_F32 (exact fp32 path).
//
// One wave computes a 32x64 tile of out = A[N x K] @ W[K x DOUT]:
//   2 row-tiles x 4 col-tiles of 16x16, 8 accumulators (64 VGPRs).
// Per k-step (K += 4): 2 x b64 A-loads + 8 x b32 B-loads feed 8 WMMAs
// (A fragment reused across 4 col-tiles, B fragment across 2 row-tiles).
//
// Fragment layouts (CDNA5 ISA 7.12.2, wave32):
//   A 16x4 f32 : lane l holds row l%16; VGPR v holds K = kbase + v + 2*(l/16)
//                -> per lane a contiguous float2 at A[row*K + kbase + 2*(l/16)]
//   B 4x16 f32 : lane l holds col l%16; VGPR v holds K = kbase + v + 2*(l/16)
//                -> two scalar loads strided by DOUT
//   C/D 16x16  : lane l col l%16; VGPR v row = rowbase + v + 8*(l/16)
// ---------------------------------------------------------------------------
template<int K, int DOUT, bool BN>
__global__ __launch_bounds__(128)
void gemm_bn_relu(const float* __restrict__ A,
                  const float* __restrict__ W,
                  const float* __restrict__ bias,
                  const float* __restrict__ gamma,
                  const float* __restrict__ beta,
                  const float* __restrict__ mean,
                  const float* __restrict__ var,
                  float* __restrict__ out) {
  const int lane    = threadIdx.x & 31;
  const int wave    = threadIdx.x >> 5;
  const int nwaves  = blockDim.x >> 5;
  const int rowbase = blockIdx.x << 5;                       // 32 rows / wave
  const int colbase = ((blockIdx.y * nwaves + wave)) << 6;   // 64 cols / wave
  const int hi      = lane >> 4;       // 0 or 1 (lane half)
  const int ln      = lane & 15;
  const int row     = rowbase + ln;    // A row (tile 0) for this lane
  const int col     = colbase + ln;    // B/C/D base column for this lane

  const float* aptr0 = A + (size_t)row * K + (hi << 1);
  const float* aptr1 = aptr0 + (size_t)16 * K;               // row + 16
  const float* wptr  = W + (size_t)(hi << 1) * DOUT + col;

  v8f c0[4] = {}, c1[4] = {};
#pragma unroll
  for (int k = 0; k < K; k += 4) {
    v2f a0 = *(const v2f*)(aptr0 + k);           // rows rowbase..+15
    v2f a1 = *(const v2f*)(aptr1 + k);           // rows rowbase+16..+31
#pragma unroll
    for (int j = 0; j < 4; ++j) {
      v2f b;
      b.x = wptr[(size_t)k * DOUT + 16 * j];             // K = k   + 2*hi
      b.y = wptr[(size_t)(k + 1) * DOUT + 16 * j];       // K = k+1 + 2*hi
      c0[j] = __builtin_amdgcn_wmma_f32_16x16x4_f32(
          false, a0, false, b, (short)0, c0[j], false, false);
      c1[j] = __builtin_amdgcn_wmma_f32_16x16x4_f32(
          false, a1, false, b, (short)0, c1[j], false, false);
    }
  }

  // epilogue: out = relu(((acc + bias) - m) * (g * rsqrt(v+eps)) + beta)
#pragma unroll
  for (int j = 0; j < 4; ++j) {
    const int cj = col + 16 * j;
    const float bs = bias[cj];
    float sc = 1.0f, sh = 0.0f;
    if (BN) {
      sc = gamma[cj] * rsqrtf(var[cj] + BN_EPS);
      sh = beta[cj] - mean[cj] * sc;
    }
    float* o0 = out + (size_t)(rowbase + (hi << 3)) * DOUT + cj;
    float* o1 = o0 + (size_t)16 * DOUT;
#pragma unroll
    for (int v = 0; v < 8; ++v) {
      float x0 = c0[j][v] + bs;
      float x1 = c1[j][v] + bs;
      if (BN) {
        x0 = fmaxf(fmaf(x0, sc, sh), 0.0f);
        x1 = fmaxf(fmaf(x1, sc, sh), 0.0f);
      }
      o0[(size_t)v * DOUT] = x0;                 // row = rowbase    + 8*hi + v
      o1[(size_t)v * DOUT] = x1;                 // row = rowbase+16 + 8*hi + v
    }
  }
}

// ---------------------------------------------------------------------------
// Host launcher
// ---------------------------------------------------------------------------
extern "C" void kernel_launch(void* const* d_in, const int* in_sizes, int n_in,
                              void* d_out, int out_size, void* d_ws,
                              size_t ws_size, hipStream_t stream) {
  const float*     x   = (const float*)d_in[0];
  const long long* ei  = (const long long*)d_in[1];   // [2, E] int64
  const float* W1a = (const float*)d_in[2];
  const float* b1a = (const float*)d_in[3];
  const float* g1a = (const float*)d_in[4];
  const float* be1a= (const float*)d_in[5];
  const float* m1a = (const float*)d_in[6];
  const float* v1a = (const float*)d_in[7];
  const float* W1b = (const float*)d_in[8];
  const float* b1b = (const float*)d_in[9];
  const float* g1b = (const float*)d_in[10];
  const float* be1b= (const float*)d_in[11];
  const float* m1b = (const float*)d_in[12];
  const float* v1b = (const float*)d_in[13];
  const float* W2a = (const float*)d_in[14];
  const float* b2a = (const float*)d_in[15];
  const float* g2a = (const float*)d_in[16];
  const float* be2a= (const float*)d_in[17];
  const float* m2a = (const float*)d_in[18];
  const float* v2a = (const float*)d_in[19];
  const float* W2b = (const float*)d_in[20];
  const float* b2b = (const float*)d_in[21];
  const float* g2b = (const float*)d_in[22];
  const float* be2b= (const float*)d_in[23];
  const float* m2b = (const float*)d_in[24];
  const float* v2b = (const float*)d_in[25];
  const float* W2c = (const float*)d_in[26];
  const float* b2c = (const float*)d_in[27];

  const long long N = (long long)in_sizes[0] / DIN;   // 100000 (divisible by 32)
  const long long E = (long long)in_sizes[1] / 2;     // 1600000

  // workspace layout: [ h0: N*128 | bufA: N*256 | bufB: N*256 ]
  char*  ws     = (char*)d_ws;
  size_t S128   = (size_t)N * DIN * sizeof(float);
  size_t S256   = (size_t)N * DH  * sizeof(float);
  float* buf128 = (float*)ws;
  float* bufA   = (float*)(ws + S128);
  float* bufB   = (float*)(ws + S128 + S256);
  float* outp   = (float*)d_out;

  const long long* srcIdx = ei;        // edge_index[0]
  const long long* dstIdx = ei + E;    // edge_index[1]

  // ---- GIN layer 1: h0 = x + segment_sum(x[src], dst) ----
  {
    long long n4 = N * DIN / 4;
    copy_f4<<<dim3((unsigned)((n4 + 255) / 256)), dim3(256), 0, stream>>>(
        (const float4*)x, (float4*)buf128, n4);
    long long total = E * (long long)(DIN / 4);
    scatter_add4<DIN><<<dim3((unsigned)((total + 255) / 256)), dim3(256), 0,
                        stream>>>(x, buf128, srcIdx, dstIdx, E);
  }

  dim3 blkH(128);                              // 4 waves x 64 cols = 256 cols
  dim3 gH((unsigned)(N / 32), 1);              // 3125 x 1
  dim3 blkO(32);                               // 1 wave covers 64 cols
  dim3 gO((unsigned)(N / 32), 1);

  // MLP 1: 128->256->256 with BN+ReLU
  gemm_bn_relu<DIN, DH, true><<<gH, blkH, 0, stream>>>(
      buf128, W1a, b1a, g1a, be1a, m1a, v1a, bufA);
  gemm_bn_relu<DH, DH, true><<<gH, blkH, 0, stream>>>(
      bufA, W1b, b1b, g1b, be1b, m1b, v1b, bufB);

  // ---- GIN layer 2: h2 = h1b + segment_sum(h1b[src], dst) (into bufA) ----
  {
    long long n4 = N * DH / 4;
    copy_f4<<<dim3((unsigned)((n4 + 255) / 256)), dim3(256), 0, stream>>>(
        (const float4*)bufB, (float4*)bufA, n4);
    long long total = E * (long long)(DH / 4);
    scatter_add4<DH><<<dim3((unsigned)((total + 255) / 256)), dim3(256), 0,
                       stream>>>(bufB, bufA, srcIdx, dstIdx, E);
  }

  // MLP 2: 256->256->256 with BN+ReLU, then 256->64 (bias only)
  gemm_bn_relu<DH, DH, true><<<gH, blkH, 0, stream>>>(
      bufA, W2a, b2a, g2a, be2a, m2a, v2a, bufB);
  gemm_bn_relu<DH, DH, true><<<gH, blkH, 0, stream>>>(
      bufB, W2b, b2b, g2b, be2b, m2b, v2b, bufA);
  gemm_bn_relu<DH, DOUTF, false><<<gO, blkO, 0, stream>>>(
      bufA, W2c, b2c, nullptr, nullptr, nullptr, nullptr, outp);
}